// MeanAggregator_24223615549924
// MI455X (gfx1250) — compile-verified
//
#include <hip/hip_runtime.h>
#include <stdint.h>

#define D 128
#define EDGES_PER_BLOCK 8   // 256 threads = 8 wave32 waves; one wave handles one edge row

__global__ __launch_bounds__(256) void zero_f4_kernel(float4* __restrict__ p, int n4) {
  int i = blockIdx.x * blockDim.x + threadIdx.x;
  if (i < n4) p[i] = make_float4(0.f, 0.f, 0.f, 0.f);
}

__global__ __launch_bounds__(256) void scatter_add_kernel(
    const float* __restrict__ msg, const int* __restrict__ index,
    float* __restrict__ sums, float* __restrict__ counts, int E)
{
  __shared__ float tile[EDGES_PER_BLOCK * D];   // 4 KB staging (8 rows x 512 B)

  const int lane        = threadIdx.x & 31;
  const int waveInBlock = threadIdx.x >> 5;
  const long long edge  = (long long)blockIdx.x * EDGES_PER_BLOCK + waveInBlock;
  if (edge >= E) return;            // wave-uniform: surviving waves have EXEC all-1s

  // ---- CDNA5 async global->LDS copy: each lane moves 16 B, wave moves 512 B row.
  // Non-temporal hint: keep the 256 MB msg stream from evicting the L2-resident
  // accumulators (sums+counts fit easily in the 192 MB L2).
  const unsigned gOff   = (unsigned)(edge * (D * 4)) + (unsigned)(lane * 16);
  const unsigned ldsOff = (unsigned)(uintptr_t)&tile[waveInBlock * D + lane * 4];
  asm volatile("global_load_async_to_lds_b128 %0, %1, %2 th:TH_LOAD_NT"
               :: "v"(ldsOff), "v"(gOff), "s"(msg)
               : "memory");

  // Prefetch the msg stream ~1 MB ahead while the async copy is in flight.
  const long long pfByte = (long long)gOff + (1ll << 20);
  if (pfByte < (long long)E * (D * 4))
    __builtin_prefetch((const char*)msg + pfByte, 0, 0);

  const int node = index[edge];     // wave-uniform load

  asm volatile("s_wait_asynccnt 0" ::: "memory");   // own wave's ASYNCcnt only

  const float4 v = *(const float4*)&tile[waveInBlock * D + lane * 4];
  float* dst = sums + (size_t)node * D + lane * 4;
  // Native fp32 atomics resolve at L2 (accumulators are L2-resident).
  unsafeAtomicAdd(dst + 0, v.x);
  unsafeAtomicAdd(dst + 1, v.y);
  unsafeAtomicAdd(dst + 2, v.z);
  unsafeAtomicAdd(dst + 3, v.w);
  if (lane == 0) unsafeAtomicAdd(counts + node, 1.0f);
}

__global__ __launch_bounds__(256) void finalize_kernel(
    float* __restrict__ out, const float* __restrict__ counts, int n4 /* = N*D/4 */)
{
  int i = blockIdx.x * blockDim.x + threadIdx.x;
  if (i >= n4) return;
  int node = i >> 5;                // 32 float4 per node row (D = 128)
  float c = counts[node];
  c = (c < 1.0f) ? 1.0f : c;        // clamp like reference
  float4* p = (float4*)out + i;
  float4 v = *p;
  v.x /= c; v.y /= c; v.z /= c; v.w /= c;
  *p = v;
}

extern "C" void kernel_launch(void* const* d_in, const int* in_sizes, int n_in,
                              void* d_out, int out_size, void* d_ws, size_t ws_size,
                              hipStream_t stream) {
  const float* msg   = (const float*)d_in[0];
  const int*   index = (const int*)d_in[1];
  // d_in[2] = t (unused by reference), d_in[3] = dim_size scalar (N = out_size/D)
  const int E = in_sizes[1];
  const int N = out_size / D;

  float* sums   = (float*)d_out;    // accumulate directly into the output buffer
  float* counts = (float*)d_ws;     // N floats of scratch

  // Zero accumulators every call (harness poisons once and never re-poisons).
  {
    int n4 = out_size / 4;
    zero_f4_kernel<<<(n4 + 255) / 256, 256, 0, stream>>>((float4*)sums, n4);
    int c4 = (N + 3) / 4;
    zero_f4_kernel<<<(c4 + 255) / 256, 256, 0, stream>>>((float4*)counts, c4);
  }

  scatter_add_kernel<<<(E + EDGES_PER_BLOCK - 1) / EDGES_PER_BLOCK, 256, 0, stream>>>(
      msg, index, sums, counts, E);

  int n4 = out_size / 4;
  finalize_kernel<<<(n4 + 255) / 256, 256, 0, stream>>>(sums, counts, n4);
}